// GaranAttention_38233798869826
// MI455X (gfx1250) — compile-verified
//
#include <hip/hip_runtime.h>
#include <hip/hip_bf16.h>

// ---------------------------------------------------------------------------
// GARAN attention, algebraically collapsed to a bandwidth-bound pipeline.
// B=32, DQ=768, T=20, DV=DO=256, H=W=64, HW=4096, temp=sqrt(256)=16.
// ---------------------------------------------------------------------------

#define Bsz   32
#define DQ    768
#define Tt    20
#define DV    256
#define DO    256
#define HW    4096
#define NTOT  (Bsz * HW)          // 131072 elements per channel for BN stats

typedef float v2f __attribute__((ext_vector_type(2)));
typedef float v8f __attribute__((ext_vector_type(8)));

// ---------------------------------------------------------------------------
// Generic 16x16 tile f32 WMMA GEMM:  C[M,N] = A[M,K] * B + bias
//   TRANSB=true : B(k,n) = W[n*ldb + k]   (PyTorch Linear weight [N,K])
//   TRANSB=false: B(k,n) = W[k*ldb + n]
// One wave (32 threads) per 16x16 output tile.  Uses V_WMMA_F32_16X16X4_F32.
// A layout (ISA 7.12.2): lane<16 -> M=lane, (v0,v1)=(K+0,K+1);
//                        lane>=16 -> M=lane-16, (v0,v1)=(K+2,K+3).
// B layout symmetric with lane = N.  D: lanes 0-15 M=0..7, lanes16-31 M=8..15.
// ---------------------------------------------------------------------------
template <bool TRANSB>
__global__ void wmma_gemm_f32(const float* __restrict__ A,
                              const float* __restrict__ Bm,
                              const float* __restrict__ bias,
                              float* __restrict__ C,
                              int K, int lda, int ldb, int ldc) {
    const int lane  = threadIdx.x;          // 0..31, one full wave32
    const int half  = lane >> 4;            // 0 -> K+0/1, 1 -> K+2/3
    const int l16   = lane & 15;
    const int nbase = blockIdx.x * 16;
    const int mbase = blockIdx.y * 16;

    v8f acc = {};
    const float* arow = A + (size_t)(mbase + l16) * lda;
    const float* brow = TRANSB ? (Bm + (size_t)(nbase + l16) * ldb) : Bm;

    for (int k = 0; k < K; k += 4) {
        const int ka = k + 2 * half;
        v2f a, b;
        a.x = arow[ka];
        a.y = arow[ka + 1];
        if (TRANSB) {
            b.x = brow[ka];
            b.y = brow[ka + 1];
        } else {
            b.x = Bm[(size_t)ka * ldb + nbase + l16];
            b.y = Bm[(size_t)(ka + 1) * ldb + nbase + l16];
        }
        acc = __builtin_amdgcn_wmma_f32_16x16x4_f32(false, a, false, b,
                                                    (short)0, acc, false, false);
    }

    const float bs   = bias ? bias[nbase + l16] : 0.0f;
    const int   mrow = mbase + half * 8;
    const int   ncol = nbase + l16;
#pragma unroll
    for (int r = 0; r < 8; ++r)
        C[(size_t)(mrow + r) * ldc + ncol] = acc[r] + bs;
}

// ---------------------------------------------------------------------------
// K0: time-average of concat([l_feats, pooler x T]) over 2T
// m[b,c] = (sum_t l_feats[b,c,t] + T * pooler[b,c]) / (2T)
// ---------------------------------------------------------------------------
__global__ void fuse_mean_kernel(const float* __restrict__ l_feats,
                                 const float* __restrict__ l_pooler,
                                 float* __restrict__ M) {
    const int idx = blockIdx.x * blockDim.x + threadIdx.x;   // b*DQ + c
    const float* p = l_feats + (size_t)idx * Tt;
    float s = 0.0f;
#pragma unroll
    for (int t = 0; t < Tt; ++t) s += p[t];
    s += (float)Tt * l_pooler[idx];
    M[idx] = s * (1.0f / (2.0f * Tt));
}

// ---------------------------------------------------------------------------
// offC[b] = q[b].b_kc, offD[b] = q[b].b_kd   (constant logit offsets)
// ---------------------------------------------------------------------------
__global__ void qoff_kernel(const float* __restrict__ Q,
                            const float* __restrict__ b_kc,
                            const float* __restrict__ b_kd,
                            float* __restrict__ offCD) {
    __shared__ float r1[256], r2[256];
    const int b = blockIdx.x, tid = threadIdx.x;
    const float q = Q[b * DO + tid];
    r1[tid] = q * b_kc[tid];
    r2[tid] = q * b_kd[tid];
    __syncthreads();
    for (int s = 128; s > 0; s >>= 1) {
        if (tid < s) { r1[tid] += r1[tid + s]; r2[tid] += r2[tid + s]; }
        __syncthreads();
    }
    if (tid == 0) { offCD[2 * b] = r1[0]; offCD[2 * b + 1] = r2[0]; }
}

__global__ void zero_kernel(float* __restrict__ p, int n) {
    const int i = blockIdx.x * blockDim.x + threadIdx.x;
    if (i < n) p[i] = 0.0f;
}

// ---------------------------------------------------------------------------
// K4: per-pixel 256-dim dots against qkc/qkd  (v read #1, coalesced over n)
// Lc[b,n]   = (vf[b,n].qkc[b] + offC[b]) / 16
// gate[b,n] = sigmoid((vf[b,n].qkd[b] + offD[b]) / 16)
// grid: 512 blocks (32 b x 16 tiles of 256 pixels), block 256
// ---------------------------------------------------------------------------
__global__ void logits_kernel(const float* __restrict__ v,
                              const float* __restrict__ QKc,
                              const float* __restrict__ QKd,
                              const float* __restrict__ offCD,
                              float* __restrict__ Lc,
                              float* __restrict__ gate) {
    __shared__ float skc[DV], skd[DV];
    const int b   = blockIdx.x >> 4;
    const int n0  = (blockIdx.x & 15) << 8;
    const int tid = threadIdx.x;
    skc[tid] = QKc[b * DV + tid];
    skd[tid] = QKd[b * DV + tid];
    __syncthreads();

    const int n = n0 + tid;
    const float* vb = v + (size_t)b * DV * HW + n;
    float sc = 0.0f, sd = 0.0f;
    for (int c = 0; c < DV; ++c) {
        if ((c & 15) == 0 && c + 16 < DV)
            __builtin_prefetch(&vb[(size_t)(c + 16) * HW], 0, 0);  // global_prefetch_b8
        const float val = vb[(size_t)c * HW];
        sc = fmaf(val, skc[c], sc);
        sd = fmaf(val, skd[c], sd);
    }
    const float lc = (sc + offCD[2 * b]) * 0.0625f;
    const float ld = (sd + offCD[2 * b + 1]) * 0.0625f;
    Lc[b * HW + n]   = lc;
    gate[b * HW + n] = 1.0f / (1.0f + __expf(-ld));
}

// ---------------------------------------------------------------------------
// Block reduction helpers (uniform control flow, one wave-group of 256)
// ---------------------------------------------------------------------------
__device__ __forceinline__ float block_sum(float x, float* red, int tid) {
    red[tid] = x; __syncthreads();
    for (int s = 128; s > 0; s >>= 1) {
        if (tid < s) red[tid] += red[tid + s];
        __syncthreads();
    }
    const float r = red[0]; __syncthreads();
    return r;
}
__device__ __forceinline__ float block_max(float x, float* red, int tid) {
    red[tid] = x; __syncthreads();
    for (int s = 128; s > 0; s >>= 1) {
        if (tid < s) red[tid] = fmaxf(red[tid], red[tid + s]);
        __syncthreads();
    }
    const float r = red[0]; __syncthreads();
    return r;
}

// ---------------------------------------------------------------------------
// K5: per-batch softmax stats + gate moments
// stats[b]=max, stats[32+b]=sum(exp), stats[64+b]=sum(gate), stats[96+b]=sum(gate^2)
// ---------------------------------------------------------------------------
__global__ void softmax_stats_kernel(const float* __restrict__ Lc,
                                     const float* __restrict__ gate,
                                     float* __restrict__ stats) {
    __shared__ float red[256];
    const int b = blockIdx.x, tid = threadIdx.x;
    const float* lcb = Lc + b * HW;
    const float* gb  = gate + b * HW;

    float lv[16], m = -1e30f;
#pragma unroll
    for (int k = 0; k < 16; ++k) { lv[k] = lcb[tid + k * 256]; m = fmaxf(m, lv[k]); }
    const float bmax = block_max(m, red, tid);

    float se = 0.0f, gs = 0.0f, gss = 0.0f;
#pragma unroll
    for (int k = 0; k < 16; ++k) {
        se += __expf(lv[k] - bmax);
        const float g = gb[tid + k * 256];
        gs += g; gss = fmaf(g, g, gss);
    }
    const float tse  = block_sum(se, red, tid);
    const float tgs  = block_sum(gs, red, tid);
    const float tgss = block_sum(gss, red, tid);
    if (tid == 0) {
        stats[b] = bmax; stats[32 + b] = tse;
        stats[64 + b] = tgs; stats[96 + b] = tgss;
    }
}

// ---------------------------------------------------------------------------
// K6: fused reductions over v (v read #2, float4):
//   S[b,d]  = sum_n softmax_w[b,n] * v[b,d,n]
//   Ct[b,d] = sum_n gate[b,tau(n)] * v[b,d,n]   (tau = spatial transpose)
//   Vsum[d] += sum_{b,n} v ;  Vsq[d] += sum v^2   (f32 global atomics)
// one block per (b,d); 256 threads x 4 float4
// ---------------------------------------------------------------------------
__global__ void reduce_pass_kernel(const float* __restrict__ v,
                                   const float* __restrict__ Lc,
                                   const float* __restrict__ gate,
                                   const float* __restrict__ stats,
                                   float* __restrict__ S,
                                   float* __restrict__ Ct,
                                   float* __restrict__ Vsum,
                                   float* __restrict__ Vsq) {
    __shared__ float red[256];
    const int bd = blockIdx.x;
    const int b  = bd >> 8;
    const int d  = bd & 255;
    const int tid = threadIdx.x;

    const float bmax = stats[b];
    const float rsum = 1.0f / stats[32 + b];

    const float4* v4  = (const float4*)(v + (size_t)bd * HW);
    const float4* lc4 = (const float4*)(Lc + (size_t)b * HW);
    const float*  gb  = gate + (size_t)b * HW;

    float as = 0.0f, ac = 0.0f, av = 0.0f, aq = 0.0f;
#pragma unroll
    for (int k = 0; k < 4; ++k) {
        const int idx = tid + k * 256;
        const float4 val = v4[idx];
        const float4 lc  = lc4[idx];
        const int n  = idx * 4;
        const int c0 = n & 63;            // column group (stays in one 64 block)
        const int r0 = n >> 6;            // row
        const float g0 = gb[(c0 + 0) * 64 + r0];
        const float g1 = gb[(c0 + 1) * 64 + r0];
        const float g2 = gb[(c0 + 2) * 64 + r0];
        const float g3 = gb[(c0 + 3) * 64 + r0];
        as += __expf(lc.x - bmax) * val.x + __expf(lc.y - bmax) * val.y +
              __expf(lc.z - bmax) * val.z + __expf(lc.w - bmax) * val.w;
        ac += g0 * val.x + g1 * val.y + g2 * val.z + g3 * val.w;
        av += val.x + val.y + val.z + val.w;
        aq += val.x * val.x + val.y * val.y + val.z * val.z + val.w * val.w;
    }
    as *= rsum;

    const float ts = block_sum(as, red, tid);
    const float tc = block_sum(ac, red, tid);
    const float tv = block_sum(av, red, tid);
    const float tq = block_sum(aq, red, tid);
    if (tid == 0) {
        S[bd]  = ts;
        Ct[bd] = tc;
        atomicAdd(&Vsum[d], tv);
        atomicAdd(&Vsq[d],  tq);
    }
}

// ---------------------------------------------------------------------------
// K8: analytic BatchNorm scale/shift per channel
// mean_d = (sum_b attn*Gs + Vsum)/N ; E[o^2] = (sum_b attn^2*Gss + 2 attn*Ct + Vsq)/N
// ---------------------------------------------------------------------------
__global__ void bnstats_kernel(const float* __restrict__ attn,
                               const float* __restrict__ Ct,
                               const float* __restrict__ stats,
                               const float* __restrict__ Vsum,
                               const float* __restrict__ Vsq,
                               const float* __restrict__ gamma,
                               const float* __restrict__ beta,
                               float* __restrict__ scale,
                               float* __restrict__ shift) {
    const int d = threadIdx.x;
    float sm = 0.0f, sq = 0.0f;
    for (int b = 0; b < Bsz; ++b) {
        const float a = attn[b * DO + d];
        sm = fmaf(a, stats[64 + b], sm);
        sq += a * a * stats[96 + b] + 2.0f * a * Ct[b * DO + d];
    }
    const float invN = 1.0f / (float)NTOT;
    const float mean = (sm + Vsum[d]) * invN;
    const float esq  = (sq + Vsq[d]) * invN;
    const float var  = esq - mean * mean;
    const float rstd = rsqrtf(var + 1e-5f);
    const float sc   = gamma[d] * rstd;
    scale[d] = sc;
    shift[d] = fmaf(-mean, sc, beta[d]);
}

// ---------------------------------------------------------------------------
// K9: out = leakyrelu( (gate[tau(n)]*attn + v) * scale + shift )
// (v read #3 + only output write; float4 B128 traffic)
// ---------------------------------------------------------------------------
__global__ void final_kernel(const float* __restrict__ v,
                             const float* __restrict__ gate,
                             const float* __restrict__ attn,
                             const float* __restrict__ scale,
                             const float* __restrict__ shift,
                             float* __restrict__ out) {
    const int bd = blockIdx.x;
    const int b  = bd >> 8;
    const int d  = bd & 255;
    const int tid = threadIdx.x;
    const float a  = attn[bd];
    const float sc = scale[d];
    const float sh = shift[d];
    const float4* v4 = (const float4*)(v + (size_t)bd * HW);
    float4*       o4 = (float4*)(out + (size_t)bd * HW);
    const float*  gb = gate + (size_t)b * HW;
#pragma unroll
    for (int k = 0; k < 4; ++k) {
        const int idx = tid + k * 256;
        float4 val = v4[idx];
        const int n  = idx * 4;
        const int c0 = n & 63;
        const int r0 = n >> 6;
        float o0 = fmaf(fmaf(gb[(c0 + 0) * 64 + r0], a, val.x), sc, sh);
        float o1 = fmaf(fmaf(gb[(c0 + 1) * 64 + r0], a, val.y), sc, sh);
        float o2 = fmaf(fmaf(gb[(c0 + 2) * 64 + r0], a, val.z), sc, sh);
        float o3 = fmaf(fmaf(gb[(c0 + 3) * 64 + r0], a, val.w), sc, sh);
        val.x = (o0 >= 0.0f) ? o0 : 0.1f * o0;
        val.y = (o1 >= 0.0f) ? o1 : 0.1f * o1;
        val.z = (o2 >= 0.0f) ? o2 : 0.1f * o2;
        val.w = (o3 >= 0.0f) ? o3 : 0.1f * o3;
        o4[idx] = val;
    }
}

// ---------------------------------------------------------------------------
// Launch
// ---------------------------------------------------------------------------
extern "C" void kernel_launch(void* const* d_in, const int* in_sizes, int n_in,
                              void* d_out, int out_size, void* d_ws, size_t ws_size,
                              hipStream_t stream) {
    (void)in_sizes; (void)n_in; (void)out_size; (void)ws_size;

    const float* l_feats  = (const float*)d_in[0];
    const float* l_pooler = (const float*)d_in[1];
    const float* v        = (const float*)d_in[2];
    const float* W_cas    = (const float*)d_in[3];
    const float* b_cas    = (const float*)d_in[4];
    const float* W_q      = (const float*)d_in[5];
    const float* b_q      = (const float*)d_in[6];
    const float* W_kc     = (const float*)d_in[7];
    const float* b_kc     = (const float*)d_in[8];
    const float* W_kd     = (const float*)d_in[9];
    const float* b_kd     = (const float*)d_in[10];
    const float* W_vs     = (const float*)d_in[11];
    const float* b_vs     = (const float*)d_in[12];
    const float* gamma    = (const float*)d_in[13];
    const float* beta     = (const float*)d_in[14];
    float* out = (float*)d_out;

    // workspace layout (floats)
    float* ws    = (float*)d_ws;
    float* Mbuf  = ws;                 // 32*768
    float* Lbuf  = Mbuf  + 24576;      // 32*768
    float* Qbuf  = Lbuf  + 24576;      // 32*256
    float* QKc   = Qbuf  + 8192;       // 32*256
    float* QKd   = QKc   + 8192;       // 32*256
    float* offCD = QKd   + 8192;       // 64
    float* Lc    = offCD + 64;         // 32*4096
    float* gate  = Lc    + 131072;     // 32*4096
    float* stats = gate  + 131072;     // 128
    float* Sbuf  = stats + 128;        // 32*256
    float* Ct    = Sbuf  + 8192;       // 32*256
    float* Vsum  = Ct    + 8192;       // 256
    float* Vsq   = Vsum  + 256;        // 256
    float* attn  = Vsq   + 256;        // 32*256
    float* scale = attn  + 8192;       // 256
    float* shift = scale + 256;        // 256

    // K0: time-averaged fused language features
    fuse_mean_kernel<<<96, 256, 0, stream>>>(l_feats, l_pooler, Mbuf);

    // WMMA projections (M=32 rows -> 2 tiles of 16)
    wmma_gemm_f32<true ><<<dim3(48, 2), 32, 0, stream>>>(Mbuf, W_cas, b_cas, Lbuf,
                                                         DQ, DQ, DQ, DQ);          // L = M Wcas^T + b
    wmma_gemm_f32<true ><<<dim3(16, 2), 32, 0, stream>>>(Lbuf, W_q, b_q, Qbuf,
                                                         DQ, DQ, DQ, DO);          // Q = L Wq^T + b
    wmma_gemm_f32<false><<<dim3(16, 2), 32, 0, stream>>>(Qbuf, W_kc, nullptr, QKc,
                                                         DO, DO, DV, DV);          // QKc = Q Wkc
    wmma_gemm_f32<false><<<dim3(16, 2), 32, 0, stream>>>(Qbuf, W_kd, nullptr, QKd,
                                                         DO, DO, DV, DV);          // QKd = Q Wkd

    // constant logit offsets q.b_kc / q.b_kd
    qoff_kernel<<<32, 256, 0, stream>>>(Qbuf, b_kc, b_kd, offCD);

    // zero the atomically-accumulated per-channel moments
    zero_kernel<<<2, 256, 0, stream>>>(Vsum, 512);

    // v pass #1: logits + gates
    logits_kernel<<<512, 256, 0, stream>>>(v, QKc, QKd, offCD, Lc, gate);

    // per-batch softmax + gate moments
    softmax_stats_kernel<<<32, 256, 0, stream>>>(Lc, gate, stats);

    // v pass #2: attention-weighted sum, BN cross term, per-channel moments
    reduce_pass_kernel<<<Bsz * DO, 256, 0, stream>>>(v, Lc, gate, stats,
                                                     Sbuf, Ct, Vsum, Vsq);

    // attn = S Wvs^T + b_vs  (WMMA)
    wmma_gemm_f32<true ><<<dim3(16, 2), 32, 0, stream>>>(Sbuf, W_vs, b_vs, attn,
                                                         DV, DV, DV, DO);

    // analytic BN scale/shift
    bnstats_kernel<<<1, 256, 0, stream>>>(attn, Ct, stats, Vsum, Vsq,
                                          gamma, beta, scale, shift);

    // v pass #3: fused residual + BN + LeakyReLU, write output
    final_kernel<<<Bsz * DO, 256, 0, stream>>>(v, gate, attn, scale, shift, out);
}